// TINY_YOLO_LOSS_68083821576348
// MI455X (gfx1250) — compile-verified
//
#include <hip/hip_runtime.h>

typedef float v2f __attribute__((ext_vector_type(2)));
typedef float v8f __attribute__((ext_vector_type(8)));

#define IMGF 416.0f

// _nls(x) = min(softplus(-x), 100) ; clamp makes the naive form exact at extremes.
__device__ __forceinline__ float nls_pos(float x) {           // _nls(x)
    return fminf(__logf(1.0f + __expf(-x)), 100.0f);
}
__device__ __forceinline__ float nls_neg(float x) {           // _nls(-x)
    return fminf(__logf(1.0f + __expf(x)), 100.0f);
}

__global__ __launch_bounds__(32)
void yolo_task_kernel(const float* __restrict__ Y1, const float* __restrict__ Y2,
                      const float* __restrict__ T, float* __restrict__ ws, int B)
{
    const int task = blockIdx.x;            // task = b*6 + (s*3 + n)
    const int lane = threadIdx.x;
    const int b = task / 6;
    const int r = task - b * 6;
    const int s = r / 3;
    const int n = r - s * 3;
    const int N  = (s == 0) ? 13 : 26;
    const int NN = N * N;
    const float invNN = 1.0f / (float)NN;
    const float* __restrict__ Y = (s == 0) ? Y1 : Y2;

    // ---- per-batch scalars (uniform across the wave) -------------------------
    const float tx = T[b*5+0], ty = T[b*5+1], tw = T[b*5+2], th = T[b*5+3];
    const int   cls = (int)T[b*5+4];

    const float ancw[6] = {40.f, 98.f, 150.f, 199.f, 225.f, 311.f};
    const float anch[6] = {40.f, 97.f, 164.f, 287.f, 184.f, 212.f};

    // IoU-based positive mask (6 anchors)
    const float twp = floorf(tw * IMGF);
    const float thp = floorf(th * IMGF);
    float iou[6];
    float maxiou = -1e30f;
#pragma unroll
    for (int a = 0; a < 6; ++a) {
        const float pw = ancw[a], ph = anch[a];
        const float I  = fminf(pw, twp) * fminf(ph, thp);
        const float A  = fmaxf(pw, twp) * fmaxf(ph, thp);
        const float P  = pw * ph;
        const float Tt = twp * thp;
        const float TP = A - P - Tt + 2.0f * I;
        const float FN = P + Tt - 2.0f * I;
        iou[a] = TP / (2.0f * TP + FN);
        maxiou = fmaxf(maxiou, iou[a]);
    }
    const float p = ((iou[r] == maxiou) || (iou[r] > 0.5f)) ? 1.0f : 0.0f;

    const float gx = tx * (float)N, gy = ty * (float)N;
    const int   cx = (int)gx,       cy = (int)gy;
    const float fx = gx - floorf(gx), fy = gy - floorf(gy);
    const float twt = __logf(tw * IMGF / ancw[r] + 1e-16f);
    const float tht = __logf(th * IMGF / anch[r] + 1e-16f);

    // ---- warm the 10 scattered cell values while we stream the map -----------
    const int    ci      = (lane < 10) ? lane : 9;                 // clamped: always in-bounds
    const size_t cellIdx = ((size_t)(b*30 + n*10 + ci) * N + cy) * N + cx;
    __builtin_prefetch(&Y[cellIdx], 0, 3);                         // global_prefetch_b8

    // ---- objectness map: sum of min(softplus(L),100) via WMMA ----------------
    // D = A x B + C with B = ones(4x16): every element of A lands in D 16x,
    // so chunk_sum = sum(D)/16. Layout-invariant => no data shuffling needed.
    const float* __restrict__ Lmap = Y + (size_t)(b*30 + n*10 + 4) * NN + lane;
    const int nfull = NN >> 6;          // full 64-elem chunks: 2 (13x13) or 10 (26x26)

    v8f acc0 = {0.f,0.f,0.f,0.f,0.f,0.f,0.f,0.f};
    v8f acc1 = {0.f,0.f,0.f,0.f,0.f,0.f,0.f,0.f};
    v2f ones; ones.x = 1.0f; ones.y = 1.0f;

    // Main loop: completely branch-free, unconditional loads, EXEC all-ones.
    int it = 0;
    for (; it + 1 < nfull; it += 2) {
        const int base = it * 64;
        const float x0 = Lmap[base +  0];
        const float x1 = Lmap[base + 32];
        const float x2 = Lmap[base + 64];
        const float x3 = Lmap[base + 96];
        v2f A0; A0.x = nls_neg(x0); A0.y = nls_neg(x1);
        v2f A1; A1.x = nls_neg(x2); A1.y = nls_neg(x3);
        acc0 = __builtin_amdgcn_wmma_f32_16x16x4_f32(
                   false, A0, false, ones, (short)0, acc0, false, false);
        acc1 = __builtin_amdgcn_wmma_f32_16x16x4_f32(
                   false, A1, false, ones, (short)0, acc1, false, false);
    }
    if (it < nfull) {                       // leftover full chunk (uniform; unused for 169/676)
        const int base = it * 64;
        const float x0 = Lmap[base +  0];
        const float x1 = Lmap[base + 32];
        v2f A; A.x = nls_neg(x0); A.y = nls_neg(x1);
        acc0 = __builtin_amdgcn_wmma_f32_16x16x4_f32(
                   false, A, false, ones, (short)0, acc0, false, false);
        ++it;
    }
    if (NN & 63) {                          // single masked tail chunk (uniform branch)
        const int i0 = nfull * 64 + lane;
        const int i1 = i0 + 32;
        // Loads are in-bounds: channels 5..9 follow the objectness map inside
        // the same (b,n) block (>= 5*NN floats of margin). Mask the INPUT so
        // out-of-range lanes compute softplus(-inf) = 0 (cheap cndmask, no branch).
        float x0 = Lmap[nfull * 64 +  0];
        float x1 = Lmap[nfull * 64 + 32];
        x0 = (i0 < NN) ? x0 : -1e30f;
        x1 = (i1 < NN) ? x1 : -1e30f;
        v2f A; A.x = nls_neg(x0); A.y = nls_neg(x1);
        acc1 = __builtin_amdgcn_wmma_f32_16x16x4_f32(
                   false, A, false, ones, (short)0, acc1, false, false);
    }

    v8f accs = acc0 + acc1;
    float msum = accs[0] + accs[1] + accs[2] + accs[3]
               + accs[4] + accs[5] + accs[6] + accs[7];
#pragma unroll
    for (int off = 16; off > 0; off >>= 1)
        msum += __shfl_xor(msum, off, 32);
    msum *= (1.0f / 16.0f);                 // every element counted 16x in D
    const float noobj = msum * invNN;

    // ---- single-cell terms: lanes 0..9 each own one channel ------------------
    const float v    = Y[cellIdx];
    const float sigv = 1.0f / (1.0f + __expf(-v));
    float coord_c = 0.0f, obj_c = 0.0f, cls_c = 0.0f;
    if (lane == 0)      { const float d = sigv - fx;  coord_c = d * d; }
    else if (lane == 1) { const float d = sigv - fy;  coord_c = d * d; }
    else if (lane == 2) { const float d = v - twt;    coord_c = d * d; }
    else if (lane == 3) { const float d = v - tht;    coord_c = d * d; }
    else if (lane == 4) { obj_c = (nls_pos(v) - nls_neg(v)) * invNN; }
    else if (lane <= 9) { cls_c = (((lane - 5) == cls) ? nls_pos(v) : nls_neg(v)) * 0.2f; }

#pragma unroll
    for (int off = 16; off > 0; off >>= 1) {
        coord_c += __shfl_xor(coord_c, off, 32);
        obj_c   += __shfl_xor(obj_c,   off, 32);
        cls_c   += __shfl_xor(cls_c,   off, 32);
    }

    if (lane == 0) {
        float* o = ws + (size_t)task * 4;
        o[0] = p * coord_c;                 // Lcoord part
        o[1] = p * (noobj + obj_c);         // Lobj part
        o[2] = p * cls_c;                   // Lcls part
        o[3] = (1.0f - p) * noobj;          // Lnoobj part
    }
}

__global__ __launch_bounds__(256)
void yolo_reduce_kernel(const float* __restrict__ ws, float* __restrict__ out, int ntasks)
{
    __shared__ float sm[256][4];
    float s0 = 0.f, s1 = 0.f, s2 = 0.f, s3 = 0.f;
    for (int t = threadIdx.x; t < ntasks; t += 256) {
        const float* o = ws + (size_t)t * 4;
        s0 += o[0]; s1 += o[1]; s2 += o[2]; s3 += o[3];
    }
    sm[threadIdx.x][0] = s0; sm[threadIdx.x][1] = s1;
    sm[threadIdx.x][2] = s2; sm[threadIdx.x][3] = s3;
    __syncthreads();
    for (int off = 128; off > 0; off >>= 1) {
        if ((int)threadIdx.x < off) {
            sm[threadIdx.x][0] += sm[threadIdx.x + off][0];
            sm[threadIdx.x][1] += sm[threadIdx.x + off][1];
            sm[threadIdx.x][2] += sm[threadIdx.x + off][2];
            sm[threadIdx.x][3] += sm[threadIdx.x + off][3];
        }
        __syncthreads();
    }
    if (threadIdx.x == 0) {
        const float Lcoord = sm[0][0];
        const float Lobj   = sm[0][1];
        const float Lcls   = sm[0][2];
        const float Lnoobj = sm[0][3];
        out[0] = Lcoord + Lcls + Lnoobj + Lobj;   // total
        out[1] = Lcoord;
        out[2] = Lcls;
        out[3] = Lnoobj;
        out[4] = Lobj;
    }
}

extern "C" void kernel_launch(void* const* d_in, const int* in_sizes, int n_in,
                              void* d_out, int out_size, void* d_ws, size_t ws_size,
                              hipStream_t stream)
{
    const float* Y1 = (const float*)d_in[0];   // (B,30,13,13) f32
    const float* Y2 = (const float*)d_in[1];   // (B,30,26,26) f32
    const float* T  = (const float*)d_in[2];   // (B,5) f32
    float* out = (float*)d_out;                // 5 f32: total, Lcoord, Lcls, Lnoobj, Lobj
    float* ws  = (float*)d_ws;

    const int B = in_sizes[2] / 5;             // 2048 in the reference
    const int ntasks = B * 6;

    yolo_task_kernel<<<ntasks, 32, 0, stream>>>(Y1, Y2, T, ws, B);
    yolo_reduce_kernel<<<1, 256, 0, stream>>>(ws, out, ntasks);
}